// RelGATTimeLayer_55121610277172
// MI455X (gfx1250) — compile-verified
//
#include <hip/hip_runtime.h>
#include <math.h>

typedef __attribute__((ext_vector_type(2))) float v2f;
typedef __attribute__((ext_vector_type(8))) float v8f;

#define N_NODES 50000
#define N_EDGES 800000
#define DIM     128
#define NHEAD   4
#define DHEAD   32
#define EDGE_IN 27   // 8 time + 16 attr + 3 dir

// ---- workspace layout (in floats) ----
#define OFF_Q   ((size_t)0)
#define OFF_K   ((size_t)(OFF_Q   + (size_t)N_NODES * DIM))
#define OFF_V   ((size_t)(OFF_K   + (size_t)N_NODES * DIM))
#define OFF_R   ((size_t)(OFF_V   + (size_t)N_NODES * DIM))
#define OFF_ACC ((size_t)(OFF_R   + (size_t)N_NODES * DIM))
#define OFF_LOG ((size_t)(OFF_ACC + (size_t)N_NODES * DIM))
#define OFF_DEN ((size_t)(OFF_LOG + (size_t)N_EDGES * NHEAD))
#define OFF_MAX ((size_t)(OFF_DEN + (size_t)N_NODES * NHEAD))

// Monotone float <-> uint encoding (order-preserving, handles negatives)
__device__ __forceinline__ unsigned encf(float f) {
  unsigned u = __float_as_uint(f);
  return (u & 0x80000000u) ? ~u : (u | 0x80000000u);
}
__device__ __forceinline__ float decf(unsigned e) {
  unsigned u = (e & 0x80000000u) ? (e & 0x7fffffffu) : ~e;
  return __uint_as_float(u);
}

// -------------------------------------------------------------------------
// 0) init: zero accumulator + denom, reset max buffer
// -------------------------------------------------------------------------
__global__ void init_kernel(float* acc, float* den, unsigned* maxbuf) {
  size_t i = (size_t)blockIdx.x * blockDim.x + threadIdx.x;
  size_t stride = (size_t)gridDim.x * blockDim.x;
  const size_t nacc = (size_t)N_NODES * DIM;
  const size_t nden = (size_t)N_NODES * NHEAD;
  for (size_t j = i; j < nacc; j += stride) acc[j] = 0.0f;
  for (size_t j = i; j < nden; j += stride) den[j] = 0.0f;
  if (i < NHEAD) maxbuf[i] = 0u;   // 0 == encoded -NaN, identity for max
}

// -------------------------------------------------------------------------
// 1) Q/K/V/res GEMMs with V_WMMA_F32_16X16X4_F32
//    one block = 16-node tile, wave w handles output columns [16w,16w+16)
// -------------------------------------------------------------------------
__global__ __launch_bounds__(256)
void qkv_res_kernel(const float* __restrict__ x,
                    const float* __restrict__ Wq, const float* __restrict__ Wk,
                    const float* __restrict__ Wv, const float* __restrict__ Wr,
                    const float* __restrict__ res_b,
                    float* __restrict__ Q, float* __restrict__ K,
                    float* __restrict__ V, float* __restrict__ R) {
  const int wave  = threadIdx.x >> 5;
  const int lane  = threadIdx.x & 31;
  const int node0 = blockIdx.x * 16;
  const int o0    = wave * 16;
  const int m16   = lane & 15;        // row-within-tile (A) / col-within-tile (B)
  const int kh    = (lane >> 4) * 2;  // K sub-offset per lane half

  // load A fragments for the whole K=128 strip: 32 steps x float2 per lane
  int row = node0 + m16;
  bool rowok = row < N_NODES;
  const float* xrow = x + (size_t)(rowok ? row : (N_NODES - 1)) * DIM;
  v2f afrag[32];
#pragma unroll
  for (int s = 0; s < 32; ++s) {
    int k = s * 4 + kh;
    float a0 = rowok ? xrow[k]     : 0.0f;
    float a1 = rowok ? xrow[k + 1] : 0.0f;
    afrag[s][0] = a0; afrag[s][1] = a1;
  }

  const float* Ws[4] = {Wq, Wk, Wv, Wr};
  float*       Ys[4] = {Q, K, V, R};

  const int dcol  = o0 + (lane & 15);
  const int rbase = node0 + ((lane >> 4) * 8);

#pragma unroll
  for (int m = 0; m < 4; ++m) {
    const float* W = Ws[m];
    const float* wrow = W + (size_t)(o0 + m16) * DIM + kh;  // B: W[o0+n, k..]
    v8f c = {};
#pragma unroll
    for (int s = 0; s < 32; ++s) {
      v2f b;
      b[0] = wrow[s * 4];
      b[1] = wrow[s * 4 + 1];
      c = __builtin_amdgcn_wmma_f32_16x16x4_f32(false, afrag[s], false, b,
                                                (short)0, c, false, false);
    }
    float bias = (m == 3) ? res_b[dcol] : 0.0f;
    float* Y = Ys[m];
#pragma unroll
    for (int r = 0; r < 8; ++r) {
      int rr = rbase + r;
      if (rr < N_NODES) Y[(size_t)rr * DIM + dcol] = c[r] + bias;
    }
  }
}

// -------------------------------------------------------------------------
// 2) per-(edge,head) logits + global max (encoded-uint atomics)
// -------------------------------------------------------------------------
__global__ __launch_bounds__(256)
void edge_logits_kernel(const int* __restrict__ eidx,
                        const float* __restrict__ etime,
                        const float* __restrict__ eattr,
                        const float* __restrict__ edir,
                        const float* __restrict__ edge_w,
                        const float* __restrict__ edge_b,
                        const float* __restrict__ attn_a,
                        const float* __restrict__ Q,
                        const float* __restrict__ Km,
                        float* __restrict__ logits,
                        unsigned* __restrict__ maxbuf) {
  __shared__ unsigned smax[NHEAD];
  if (threadIdx.x < NHEAD) smax[threadIdx.x] = 0u;
  __syncthreads();

  size_t idx = (size_t)blockIdx.x * blockDim.x + threadIdx.x;
  if (idx < (size_t)N_EDGES * NHEAD) {
    int e = (int)(idx >> 2);
    int h = (int)(idx & 3);
    int src = eidx[2 * e];
    int dst = eidx[2 * e + 1];

    // edge bias: tanh(e_feat . edge_w[h] + edge_b[h])
    const float* w = edge_w + h * EDGE_IN;
    float acc = edge_b[h];
    float t = etime[e];
#pragma unroll
    for (int k = 1; k <= 4; ++k) {
      float ang = 6.283185307179586f * t * (float)k;
      acc += sinf(ang) * w[k - 1] + cosf(ang) * w[3 + k];
    }
    const float* ea = eattr + (size_t)e * 16;
#pragma unroll
    for (int k = 0; k < 16; ++k) acc += ea[k] * w[8 + k];
    const float* ed = edir + (size_t)e * 3;
#pragma unroll
    for (int k = 0; k < 3; ++k) acc += ed[k] * w[24 + k];
    float lg = tanhf(acc);

    // attention dot: a[h, 0:32].Q[dst,h,:] + a[h, 32:64].K[src,h,:]
    const float* q  = Q  + (size_t)dst * DIM + h * DHEAD;
    const float* kk = Km + (size_t)src * DIM + h * DHEAD;
    const float* av = attn_a + h * 2 * DHEAD;
#pragma unroll 8
    for (int d = 0; d < DHEAD; ++d)
      lg += q[d] * av[d] + kk[d] * av[DHEAD + d];

    lg = (lg > 0.0f) ? lg : 0.2f * lg;     // leaky relu
    logits[idx] = lg;
    atomicMax(&smax[h], encf(lg));
  }
  __syncthreads();
  if (threadIdx.x < NHEAD) atomicMax(&maxbuf[threadIdx.x], smax[threadIdx.x]);
}

// -------------------------------------------------------------------------
// 3) exp (max-stabilized, clamped) + denom segment-sum
// -------------------------------------------------------------------------
__global__ __launch_bounds__(256)
void edge_exp_kernel(const int* __restrict__ eidx,
                     float* __restrict__ logits,
                     const unsigned* __restrict__ maxbuf,
                     float* __restrict__ den) {
  size_t idx = (size_t)blockIdx.x * blockDim.x + threadIdx.x;
  if (idx >= (size_t)N_EDGES * NHEAD) return;
  int e = (int)(idx >> 2);
  int h = (int)(idx & 3);
  float mh = decf(maxbuf[h]);
  float ex = fminf(expf(logits[idx] - mh), 1000000.0f);
  logits[idx] = ex;                       // overwrite in place with exp
  int dst = eidx[2 * e + 1];
  atomicAdd(&den[(size_t)dst * NHEAD + h], ex);
}

// -------------------------------------------------------------------------
// 4) alpha = exp/denom (write to output) + V scatter-add into accumulator
// -------------------------------------------------------------------------
__global__ __launch_bounds__(256)
void edge_msg_kernel(const int* __restrict__ eidx,
                     const float* __restrict__ logits,
                     const float* __restrict__ den,
                     const float* __restrict__ V,
                     float* __restrict__ alphas,
                     float* __restrict__ acc) {
  size_t idx = (size_t)blockIdx.x * blockDim.x + threadIdx.x;
  if (idx >= (size_t)N_EDGES * NHEAD) return;
  int e = (int)(idx >> 2);
  int h = (int)(idx & 3);
  int src = eidx[2 * e];
  int dst = eidx[2 * e + 1];
  const float* v = V + (size_t)src * DIM + h * DHEAD;
  __builtin_prefetch(v, 0, 1);            // global_prefetch_b8
  float alpha = logits[idx] / (den[(size_t)dst * NHEAD + h] + 1e-12f);
  alphas[idx] = alpha;
  float* o = acc + (size_t)dst * DIM + h * DHEAD;
#pragma unroll 8
  for (int d = 0; d < DHEAD; ++d) atomicAdd(&o[d], v[d] * alpha);
}

// -------------------------------------------------------------------------
// 5) out GEMM (WMMA) + bias + residual + LayerNorm + ReLU
// -------------------------------------------------------------------------
__global__ __launch_bounds__(256)
void out_ln_kernel(const float* __restrict__ acc,
                   const float* __restrict__ out_w,
                   const float* __restrict__ out_b,
                   const float* __restrict__ R,
                   const float* __restrict__ ln_g,
                   const float* __restrict__ ln_b,
                   float* __restrict__ y) {
  __shared__ float zt[16][DIM + 4];   // padded to dodge bank conflicts

  const int wave  = threadIdx.x >> 5;
  const int lane  = threadIdx.x & 31;
  const int node0 = blockIdx.x * 16;
  const int o0    = wave * 16;
  const int m16   = lane & 15;
  const int kh    = (lane >> 4) * 2;

  int row = node0 + m16;
  const float* arow = acc + (size_t)((row < N_NODES) ? row : (N_NODES - 1)) * DIM;
  const float* wrow = out_w + (size_t)(o0 + m16) * DIM + kh;

  v8f c = {};
#pragma unroll
  for (int s = 0; s < 32; ++s) {
    int k = s * 4 + kh;
    v2f a; a[0] = arow[k];     a[1] = arow[k + 1];
    v2f b; b[0] = wrow[s * 4]; b[1] = wrow[s * 4 + 1];
    c = __builtin_amdgcn_wmma_f32_16x16x4_f32(false, a, false, b,
                                              (short)0, c, false, false);
  }

  // z = out + out_b + res  -> LDS tile (local rows 0..15, cols 0..127)
  const int dcol  = o0 + (lane & 15);
  const int lr0   = (lane >> 4) * 8;
  float ob = out_b[dcol];
#pragma unroll
  for (int r = 0; r < 8; ++r) {
    int lr = lr0 + r;
    int gr = node0 + lr;
    int grs = (gr < N_NODES) ? gr : (N_NODES - 1);
    zt[lr][dcol] = c[r] + ob + R[(size_t)grs * DIM + dcol];
  }
  __syncthreads();

  // LayerNorm + ReLU: wave handles rows 2*wave, 2*wave+1
#pragma unroll
  for (int rr = 0; rr < 2; ++rr) {
    int lr = wave * 2 + rr;
    float vals[4];
    float s = 0.0f, s2 = 0.0f;
#pragma unroll
    for (int i = 0; i < 4; ++i) {
      float z = zt[lr][lane + 32 * i];
      vals[i] = z; s += z; s2 += z * z;
    }
#pragma unroll
    for (int off = 16; off > 0; off >>= 1) {
      s  += __shfl_xor(s,  off, 32);
      s2 += __shfl_xor(s2, off, 32);
    }
    float mu  = s * (1.0f / DIM);
    float var = s2 * (1.0f / DIM) - mu * mu;
    float rstd = rsqrtf(var + 1e-5f);
    int gr = node0 + lr;
    if (gr < N_NODES) {
#pragma unroll
      for (int i = 0; i < 4; ++i) {
        int col = lane + 32 * i;
        float yv = (vals[i] - mu) * rstd * ln_g[col] + ln_b[col];
        y[(size_t)gr * DIM + col] = fmaxf(yv, 0.0f);
      }
    }
  }
}

// -------------------------------------------------------------------------
// 6) tail: copy edge_index (as float) and edge_time into output
// -------------------------------------------------------------------------
__global__ void tail_kernel(const int* __restrict__ eidx,
                            const float* __restrict__ etime,
                            float* __restrict__ o_idx,
                            float* __restrict__ o_time) {
  size_t i = (size_t)blockIdx.x * blockDim.x + threadIdx.x;
  if (i < (size_t)N_EDGES * 2) o_idx[i] = (float)eidx[i];
  if (i < (size_t)N_EDGES)     o_time[i] = etime[i];
}

// -------------------------------------------------------------------------
extern "C" void kernel_launch(void* const* d_in, const int* in_sizes, int n_in,
                              void* d_out, int out_size, void* d_ws, size_t ws_size,
                              hipStream_t stream) {
  const float* x     = (const float*)d_in[0];
  const int*   eidx  = (const int*)  d_in[1];
  const float* etime = (const float*)d_in[2];
  const float* eattr = (const float*)d_in[3];
  const float* edir  = (const float*)d_in[4];
  const float* Wq    = (const float*)d_in[5];
  const float* Wk    = (const float*)d_in[6];
  const float* Wv    = (const float*)d_in[7];
  const float* ew    = (const float*)d_in[8];
  const float* eb    = (const float*)d_in[9];
  const float* av    = (const float*)d_in[10];
  const float* ow    = (const float*)d_in[11];
  const float* ob    = (const float*)d_in[12];
  const float* rw    = (const float*)d_in[13];
  const float* rb    = (const float*)d_in[14];
  const float* lng   = (const float*)d_in[15];
  const float* lnb   = (const float*)d_in[16];

  float* ws  = (float*)d_ws;
  float* Q   = ws + OFF_Q;
  float* K   = ws + OFF_K;
  float* V   = ws + OFF_V;
  float* R   = ws + OFF_R;
  float* acc = ws + OFF_ACC;
  float* lg  = ws + OFF_LOG;
  float* den = ws + OFF_DEN;
  unsigned* maxbuf = (unsigned*)(ws + OFF_MAX);

  float* y      = (float*)d_out;
  float* alphas = y + (size_t)N_NODES * DIM;
  float* o_idx  = alphas + (size_t)N_EDGES * NHEAD;
  float* o_time = o_idx + (size_t)N_EDGES * 2;

  const int ntiles = (N_NODES + 15) / 16;                 // 3125
  const int eblks  = (N_EDGES * NHEAD + 255) / 256;       // 12500

  init_kernel<<<2048, 256, 0, stream>>>(acc, den, maxbuf);
  qkv_res_kernel<<<ntiles, 256, 0, stream>>>(x, Wq, Wk, Wv, rw, rb, Q, K, V, R);
  edge_logits_kernel<<<eblks, 256, 0, stream>>>(eidx, etime, eattr, edir, ew, eb,
                                                av, Q, K, lg, maxbuf);
  edge_exp_kernel<<<eblks, 256, 0, stream>>>(eidx, lg, maxbuf, den);
  edge_msg_kernel<<<eblks, 256, 0, stream>>>(eidx, lg, den, V, alphas, acc);
  out_ln_kernel<<<ntiles, 256, 0, stream>>>(acc, ow, ob, R, lng, lnb, y);
  tail_kernel<<<(N_EDGES * 2 + 255) / 256, 256, 0, stream>>>(eidx, etime, o_idx, o_time);
}